// Wav2Frames_43508018708665
// MI455X (gfx1250) — compile-verified
//
#include <hip/hip_runtime.h>
#include <stdint.h>

// out[b, c, f] = x[b, f*WINSTEP + c]   (conv1d with identity kernel == framing)
#define WINLEN     400
#define WINSTEP    160
#define BATCH      32
#define T_LEN      480000
#define NFRAMES    2998                       // (T_LEN - WINLEN)/WINSTEP + 1
#define TF         32                         // frames per tile (power of 2)
#define NTILES     ((NFRAMES + TF - 1) / TF)  // 94
#define LDS_STRIDE 401                        // odd -> conflict-free bank access
#define NTHREADS   256                        // 8 waves of 32
#define NWAVES     (NTHREADS / 32)

typedef __attribute__((address_space(1))) int  gint_t;
typedef __attribute__((address_space(3))) int  lint_t;

__global__ __launch_bounds__(NTHREADS, 2)
void wav2frames_43508018708665_kernel(const float* __restrict__ x,
                                      float* __restrict__ out) {
    __shared__ float tile[TF * LDS_STRIDE];   // 51,328 B

    const int tid  = threadIdx.x;
    const int lane = tid & 31;
    const int wave = tid >> 5;
    const int ti   = blockIdx.x;              // frame-tile index
    const int b    = blockIdx.y;              // batch index
    const int f0   = ti * TF;
    const int nvalid = (NFRAMES - f0 < TF) ? (NFRAMES - f0) : TF;

    const float* xb = x + (size_t)b * T_LEN;

    // ---- Fill: global -> LDS (framed, padded layout) via async-to-LDS ----
    // Wave w owns frame rows fi = w, w+8, ...; lanes sweep c contiguously:
    // 32 lanes x 4B = 128B coalesced per async instruction, LDS dests
    // contiguous. No divisions; addresses are simple strided increments.
    for (int fi = wave; fi < nvalid; fi += NWAVES) {
        const float* grow = xb + (size_t)(f0 + fi) * WINSTEP;
        float* lrow = &tile[fi * LDS_STRIDE];
        for (int c = lane; c < WINLEN; c += 32) {
#if __has_builtin(__builtin_amdgcn_global_load_async_to_lds_b32)
            __builtin_amdgcn_global_load_async_to_lds_b32(
                (gint_t*)(grow + c), (lint_t*)(lrow + c),
                /*offset=*/0, /*cpol=*/0);
#else
            uint32_t lds_addr = (uint32_t)(uintptr_t)(lint_t*)(lrow + c);
            asm volatile("global_load_async_to_lds_b32 %0, %1, off"
                         :: "v"(lds_addr), "v"(grow + c) : "memory");
#endif
        }
    }

#if __has_builtin(__builtin_amdgcn_s_wait_asynccnt)
    __builtin_amdgcn_s_wait_asynccnt(0);
#else
    asm volatile("s_wait_asynccnt 0" ::: "memory");
#endif
    __syncthreads();

    // ---- Store: LDS -> global, lanes sweep the frame axis (coalesced) ----
    // Each thread owns a fixed frame fi = lane and sweeps channels c.
    // LDS bank = (fi*401 + c) mod 64 = (fi*17 + c) mod 64: 17 odd ->
    // 32 lanes hit 32 distinct banks, conflict-free. Global stores:
    // lanes = consecutive f -> 128B per wave-instruction.
    const int f = f0 + lane;                  // fi == lane
    if (f < NFRAMES) {
        float* ocol = out + (size_t)b * (size_t)WINLEN * NFRAMES + f;
        const float* lcol = &tile[lane * LDS_STRIDE];
        for (int c = wave; c < WINLEN; c += NWAVES) {
            ocol[(size_t)c * NFRAMES] = lcol[c];
        }
    }
}

extern "C" void kernel_launch(void* const* d_in, const int* in_sizes, int n_in,
                              void* d_out, int out_size, void* d_ws, size_t ws_size,
                              hipStream_t stream) {
    (void)in_sizes; (void)n_in; (void)out_size; (void)d_ws; (void)ws_size;
    const float* x  = (const float*)d_in[0];
    // d_in[1] (W = identity) and d_in[2] (winstep = 160) are fixed by setup;
    // the framing shortcut encodes them exactly.
    float* out = (float*)d_out;

    dim3 grid(NTILES, BATCH, 1);
    dim3 block(NTHREADS, 1, 1);
    wav2frames_43508018708665_kernel<<<grid, block, 0, stream>>>(x, out);
}